// TopKTopPSampler_50483045597940
// MI455X (gfx1250) — compile-verified
//
#include <hip/hip_runtime.h>
#include <hip/hip_bf16.h>

#define TPB     1024
#define NBIN    4096
#define CAPK    8192     // candidate cap per row (global ws, used for top-k and top-p)
#define TILEF   4000     // floats per async LDS tile (16000 B)
#define SLICES  4        // column slices per row for streaming kernels
#define NEGINF  (-__builtin_inff())

// gfx1250 async global->LDS path (probe-verified signature: (AS1 v4i*, AS3 v4i*, i32, i32)).
#if defined(__HIP_DEVICE_COMPILE__) && defined(__gfx1250__) && \
    __has_builtin(__builtin_amdgcn_global_load_async_to_lds_b128) && \
    __has_builtin(__builtin_amdgcn_s_wait_asynccnt)
#define USE_ASYNC_LDS 1
#else
#define USE_ASYNC_LDS 0
#endif

#if USE_ASYNC_LDS
typedef int v4i_ __attribute__((ext_vector_type(4)));
typedef __attribute__((address_space(1))) v4i_* g1_v4i;
typedef __attribute__((address_space(3))) v4i_* l3_v4i;
#define ASYNC_B128(gsrc, ldst) \
  __builtin_amdgcn_global_load_async_to_lds_b128((g1_v4i)(gsrc), (l3_v4i)(ldst), 0, 0)
#endif

// ---- monotonic float <-> uint key (ascending) ----
static __device__ __forceinline__ unsigned f2key(float f) {
  unsigned x = __float_as_uint(f);
  return x ^ ((x & 0x80000000u) ? 0xFFFFFFFFu : 0x80000000u);
}
static __device__ __forceinline__ float key2f(unsigned k) {
  unsigned x = (k & 0x80000000u) ? (k ^ 0x80000000u) : ~k;
  return __uint_as_float(x);
}

// in-LDS bitonic sort, descending, N power of two, all threads participate
static __device__ __forceinline__ void bitonic_desc(float* a, int N) {
  for (int ksz = 2; ksz <= N; ksz <<= 1) {
    for (int j = ksz >> 1; j > 0; j >>= 1) {
      for (int i = (int)threadIdx.x; i < N; i += (int)blockDim.x) {
        int ixj = i ^ j;
        if (ixj > i) {
          float x = a[i], y = a[ixj];
          bool up = (i & ksz) == 0;
          if (up ? (x < y) : (x > y)) { a[i] = y; a[ixj] = x; }
        }
      }
      __syncthreads();
    }
  }
}

// K0: zero the accumulator region (harness does not re-poison between replays).
__global__ void k_init(unsigned* __restrict__ w, long n) {
  long i = (long)blockIdx.x * blockDim.x + threadIdx.x;
  long stride = (long)gridDim.x * blockDim.x;
  for (; i < n; i += stride) w[i] = 0u;
}

// K1: sliced row max + 4096-bin count histogram (key[31:20]); merge to global.
__global__ __launch_bounds__(TPB) void k_hist(const float* __restrict__ logits, int V, int chunk,
                                              unsigned* __restrict__ ghist,
                                              unsigned* __restrict__ gmax) {
  __shared__ unsigned hist[NBIN];
  __shared__ unsigned lmax;
#if USE_ASYNC_LDS
  __shared__ float tile[2][TILEF];
#endif
  const int row = blockIdx.x, sl = blockIdx.y, tid = threadIdx.x;
  const int start = sl * chunk;
  int len = V - start; if (len > chunk) len = chunk; if (len < 0) len = 0;
  for (int i = tid; i < NBIN; i += TPB) hist[i] = 0u;
  if (tid == 0) lmax = 0u;
  __syncthreads();
  const float* base = logits + (size_t)row * (size_t)V + start;
  unsigned mk = 0u;
  int tail = 0;
#if USE_ASYNC_LDS
  if ((((size_t)base) & 15) == 0) {          // 16B alignment required for b128
    const int nt = len / TILEF;
    tail = nt * TILEF;
    if (nt > 0) {
      if (tid < TILEF / 4)
        ASYNC_B128(base + tid * 4, &tile[0][tid * 4]);
      for (int t = 0; t < nt; ++t) {
        const int cur = t & 1;
        if (t + 1 < nt) {
          if (tid < TILEF / 4)
            ASYNC_B128(base + (size_t)(t + 1) * TILEF + tid * 4, &tile[cur ^ 1][tid * 4]);
          __builtin_amdgcn_s_wait_asynccnt(1);   // oldest tile done, next in flight
        } else {
          __builtin_amdgcn_s_wait_asynccnt(0);
        }
        __syncthreads();
        for (int j = tid; j < TILEF; j += TPB) {
          const float v = tile[cur][j];
          const unsigned kx = f2key(v);
          mk = mk > kx ? mk : kx;
          atomicAdd(&hist[kx >> 20], 1u);
        }
        __syncthreads();
      }
    }
  }
#endif
  for (int i = tail + tid; i < len; i += TPB) {
    const float v = base[i];
    const unsigned kx = f2key(v);
    mk = mk > kx ? mk : kx;
    atomicAdd(&hist[kx >> 20], 1u);
  }
  atomicMax(&lmax, mk);
  __syncthreads();
  unsigned* gh = ghist + (size_t)row * NBIN;
  for (int i = tid; i < NBIN; i += TPB) {
    const unsigned c = hist[i];
    if (c) atomicAdd(&gh[i], c);   // skip empty bins: tiny atomic traffic
  }
  if (tid == 0) atomicMax(&gmax[row], lmax);
}

// K2: per-row scan of merged histogram -> top-k boundary bin + residual rank.
__global__ __launch_bounds__(TPB) void k_selK(const int* __restrict__ karr,
                                              const unsigned* __restrict__ ghist,
                                              const unsigned* __restrict__ gmax,
                                              float* __restrict__ ctlM,
                                              int* __restrict__ ctlB1,
                                              int* __restrict__ ctlKp) {
  __shared__ unsigned hist[NBIN];
  const int row = blockIdx.x, tid = threadIdx.x;
  const unsigned* gh = ghist + (size_t)row * NBIN;
  for (int i = tid; i < NBIN; i += TPB) hist[i] = gh[i];
  __syncthreads();
  if (tid == 0) {
    const unsigned k = (unsigned)karr[row];
    unsigned cum = 0u;
    int b1 = 0, kp = 1;
    for (int b = NBIN - 1; b >= 0; --b) {
      const unsigned c = hist[b];
      if (cum + c >= k) { b1 = b; kp = (int)(k - cum); break; }
      cum += c;
    }
    ctlM[row] = key2f(gmax[row]);
    ctlB1[row] = b1;
    ctlKp[row] = kp;   // 1-based rank from top within boundary bin
  }
}

// K3: compact top-k boundary-bin elements to workspace (sliced).
__global__ __launch_bounds__(TPB) void k_collectK(const float* __restrict__ logits, int V, int chunk,
                                                  const int* __restrict__ ctlB1,
                                                  int* __restrict__ ctlCnt,
                                                  float* __restrict__ candBuf) {
  const int row = blockIdx.x, sl = blockIdx.y, tid = threadIdx.x;
  const int start = sl * chunk;
  int len = V - start; if (len > chunk) len = chunk; if (len < 0) len = 0;
  const float* base = logits + (size_t)row * (size_t)V + start;
  const unsigned b1 = (unsigned)ctlB1[row];
  float* cb = candBuf + (size_t)row * CAPK;
  for (int i = tid; i < len; i += TPB) {
    const float v = base[i];
    if ((f2key(v) >> 20) == b1) {
      const int id = atomicAdd(&ctlCnt[row], 1);
      if (id < CAPK) cb[id] = v;
    }
  }
}

// K4: LDS bitonic sort of candidates -> exact k-th-largest threshold T.
__global__ __launch_bounds__(TPB) void k_selT(const int* __restrict__ ctlKp,
                                              const int* __restrict__ ctlCnt,
                                              const float* __restrict__ candBuf,
                                              float* __restrict__ ctlT) {
  __shared__ float c[CAPK];
  const int row = blockIdx.x, tid = threadIdx.x;
  int n = ctlCnt[row]; if (n > CAPK) n = CAPK;
  const float* cb = candBuf + (size_t)row * CAPK;
  for (int i = tid; i < CAPK; i += TPB) c[i] = (i < n) ? cb[i] : NEGINF;
  __syncthreads();
  bitonic_desc(c, CAPK);
  if (tid == 0) {
    int idx = ctlKp[row] - 1;
    if (idx > n - 1) idx = n - 1;
    if (idx < 0) idx = 0;
    ctlT[row] = c[idx];
  }
}

// K5: sliced Z = sum exp(v-M) over v>=T + exp-mass histogram; merge to global.
__global__ __launch_bounds__(TPB) void k_pmass(const float* __restrict__ logits, int V, int chunk,
                                               const float* __restrict__ ctlM,
                                               const float* __restrict__ ctlT,
                                               float* __restrict__ gfhist,
                                               float* __restrict__ ctlZ) {
  __shared__ float fh[NBIN];
  __shared__ float zsh;
  const int row = blockIdx.x, sl = blockIdx.y, tid = threadIdx.x;
  const int start = sl * chunk;
  int len = V - start; if (len > chunk) len = chunk; if (len < 0) len = 0;
  for (int i = tid; i < NBIN; i += TPB) fh[i] = 0.f;
  if (tid == 0) zsh = 0.f;
  __syncthreads();
  const float* base = logits + (size_t)row * (size_t)V + start;
  const float M = ctlM[row], T = ctlT[row];
  float z = 0.f;
  for (int i = tid; i < len; i += TPB) {
    const float v = base[i];
    if (v >= T) {
      const float e = __expf(v - M);
      z += e;
      atomicAdd(&fh[f2key(v) >> 20], e);   // ds_add_f32
    }
  }
  atomicAdd(&zsh, z);
  __syncthreads();
  float* gf = gfhist + (size_t)row * NBIN;
  for (int i = tid; i < NBIN; i += TPB) {
    const float m = fh[i];
    if (m != 0.f) atomicAdd(&gf[i], m);    // global_atomic_add_f32
  }
  if (tid == 0) atomicAdd(&ctlZ[row], zsh);
}

// K6: per-row scan of exp-mass histogram -> nucleus boundary bin bp + residual R.
__global__ __launch_bounds__(TPB) void k_selP(const float* __restrict__ parr,
                                              const float* __restrict__ gfhist,
                                              const float* __restrict__ ctlZ,
                                              int* __restrict__ ctlBp,
                                              float* __restrict__ ctlR) {
  __shared__ float fh[NBIN];
  const int row = blockIdx.x, tid = threadIdx.x;
  const float* gf = gfhist + (size_t)row * NBIN;
  for (int i = tid; i < NBIN; i += TPB) fh[i] = gf[i];
  __syncthreads();
  if (tid == 0) {
    const float Z = ctlZ[row];
    const float target = parr[row] * Z;    // drop v iff exp-mass strictly above v >= p*Z
    float A = 0.f, R = 0.f;
    int bp = -1;
    for (int b = NBIN - 1; b >= 0; --b) {
      const float m = fh[b];
      if (A + m >= target) { bp = b; R = target - A; break; }
      A += m;
    }
    ctlBp[row] = bp; ctlR[row] = R;
  }
}

// K7: compact nucleus boundary-bin survivors to workspace (sliced).
__global__ __launch_bounds__(TPB) void k_collectP(const float* __restrict__ logits, int V, int chunk,
                                                  const float* __restrict__ ctlT,
                                                  const int* __restrict__ ctlBp,
                                                  int* __restrict__ ctlCnt2,
                                                  float* __restrict__ candBuf) {
  const int row = blockIdx.x, sl = blockIdx.y, tid = threadIdx.x;
  const int bp = ctlBp[row];
  if (bp < 0) return;                       // uniform per block
  const int start = sl * chunk;
  int len = V - start; if (len > chunk) len = chunk; if (len < 0) len = 0;
  const float* base = logits + (size_t)row * (size_t)V + start;
  const float T = ctlT[row];
  float* cb = candBuf + (size_t)row * CAPK;
  for (int i = tid; i < len; i += TPB) {
    const float v = base[i];
    if (v >= T && (int)(f2key(v) >> 20) == bp) {
      const int id = atomicAdd(&ctlCnt2[row], 1);
      if (id < CAPK) cb[id] = v;
    }
  }
}

// K8: sort boundary-bin survivors, walk exp mass -> exact in-bin cut Tp.
__global__ __launch_bounds__(TPB) void k_resolveP(const float* __restrict__ ctlM,
                                                  const int* __restrict__ ctlBp,
                                                  const float* __restrict__ ctlR,
                                                  const int* __restrict__ ctlCnt2,
                                                  const float* __restrict__ candBuf,
                                                  float* __restrict__ ctlTp) {
  __shared__ float c[CAPK];
  const int row = blockIdx.x, tid = threadIdx.x;
  const int bp = ctlBp[row];
  if (bp < 0) { if (tid == 0) ctlTp[row] = NEGINF; return; }   // uniform exit
  int n = ctlCnt2[row]; if (n > CAPK) n = CAPK;
  const float* cb = candBuf + (size_t)row * CAPK;
  for (int i = tid; i < CAPK; i += TPB) c[i] = (i < n) ? cb[i] : NEGINF;
  __syncthreads();
  bitonic_desc(c, CAPK);
  if (tid == 0) {
    const float M = ctlM[row], R = ctlR[row];
    float S = 0.f, Tp = __builtin_inff();
    for (int i = 0; i < n; ++i) {
      if (S < R) { Tp = c[i]; S += __expf(c[i] - M); }
      else break;
    }
    ctlTp[row] = Tp;   // keep v in bin bp iff v >= Tp
  }
}

// K9: streaming masked write (sliced, float4 + prefetch).
__global__ __launch_bounds__(TPB) void k_final(const float* __restrict__ logits, int V, int chunk,
                                               const float* __restrict__ ctlM,
                                               const float* __restrict__ ctlT,
                                               const int* __restrict__ ctlBp,
                                               const float* __restrict__ ctlTp,
                                               float* __restrict__ out) {
  const int row = blockIdx.x, sl = blockIdx.y, tid = threadIdx.x;
  const int start = sl * chunk;
  int len = V - start; if (len > chunk) len = chunk; if (len < 0) len = 0;
  const float M = ctlM[row], T = ctlT[row], Tp = ctlTp[row];
  const int bp = ctlBp[row];
  const float* base = logits + (size_t)row * (size_t)V + start;
  float* obase = out + (size_t)row * (size_t)V + start;
  if (((V | chunk) & 3) == 0) {            // uniform over grid
    const int n4 = len >> 2;
    const float4* s4 = (const float4*)base;
    float4* d4 = (float4*)obase;
    for (int i = tid; i < n4; i += TPB) {
      if (i + 2048 < n4) __builtin_prefetch((const void*)(s4 + i + 2048), 0, 1);  // global_prefetch_b8
      float4 v = s4[i];
      float r[4] = {v.x, v.y, v.z, v.w};
#pragma unroll
      for (int cc = 0; cc < 4; ++cc) {
        const float x = r[cc];
        const int bin = (int)(f2key(x) >> 20);
        bool keep = (x >= T) && (bp < 0 || bin > bp || (bin == bp && x >= Tp));
        keep = keep || (x == M);           // always keep argmax
        r[cc] = keep ? x : NEGINF;
      }
      d4[i] = make_float4(r[0], r[1], r[2], r[3]);
    }
  } else {
    for (int i = tid; i < len; i += TPB) {
      const float x = base[i];
      const int bin = (int)(f2key(x) >> 20);
      bool keep = (x >= T) && (bp < 0 || bin > bp || (bin == bp && x >= Tp));
      keep = keep || (x == M);
      obase[i] = keep ? x : NEGINF;
    }
  }
}

extern "C" void kernel_launch(void* const* d_in, const int* in_sizes, int n_in,
                              void* d_out, int out_size, void* d_ws, size_t ws_size,
                              hipStream_t stream) {
  const float* logits = (const float*)d_in[0];
  const int*   karr   = (const int*)d_in[1];
  const float* parr   = (const float*)d_in[2];
  float* out = (float*)d_out;
  const int B = in_sizes[1];
  const int V = in_sizes[0] / B;
  const int chunk = (((V + SLICES - 1) / SLICES) + 3) & ~3;

  // ---- workspace layout (word units) ----
  float* wsf    = (float*)d_ws;
  float* ctlM   = wsf + 0 * B;
  float* ctlT   = wsf + 1 * B;
  float* ctlTp  = wsf + 2 * B;
  float* ctlR   = wsf + 3 * B;
  int*   ctlB1  = (int*)(wsf + 4 * B);
  int*   ctlKp  = (int*)(wsf + 5 * B);
  int*   ctlBp  = (int*)(wsf + 6 * B);
  // zeroed accumulator region:
  float*    ctlZ   = wsf + 7 * B;
  unsigned* gmax   = (unsigned*)(wsf + 8 * B);
  int*      ctlCnt = (int*)(wsf + 9 * B);
  int*      ctlCnt2= (int*)(wsf + 10 * B);
  unsigned* ghist  = (unsigned*)(wsf + 11 * B);                    // B*NBIN u32
  float*    gfhist = (float*)(ghist + (size_t)B * NBIN);           // B*NBIN f32
  const long zeroWords = 4L * B + 2L * B * NBIN;
  float*    candBuf = gfhist + (size_t)B * NBIN;                   // B*CAPK f32

  dim3 blk(TPB);
  dim3 gRow(B);
  dim3 gSlice(B, SLICES);

  k_init    <<<dim3(256), dim3(256), 0, stream>>>((unsigned*)ctlZ, zeroWords);
  k_hist    <<<gSlice, blk, 0, stream>>>(logits, V, chunk, ghist, gmax);
  k_selK    <<<gRow,   blk, 0, stream>>>(karr, ghist, gmax, ctlM, ctlB1, ctlKp);
  k_collectK<<<gSlice, blk, 0, stream>>>(logits, V, chunk, ctlB1, ctlCnt, candBuf);
  k_selT    <<<gRow,   blk, 0, stream>>>(ctlKp, ctlCnt, candBuf, ctlT);
  k_pmass   <<<gSlice, blk, 0, stream>>>(logits, V, chunk, ctlM, ctlT, gfhist, ctlZ);
  k_selP    <<<gRow,   blk, 0, stream>>>(parr, gfhist, ctlZ, ctlBp, ctlR);
  k_collectP<<<gSlice, blk, 0, stream>>>(logits, V, chunk, ctlT, ctlBp, ctlCnt2, candBuf);
  k_resolveP<<<gRow,   blk, 0, stream>>>(ctlM, ctlBp, ctlR, ctlCnt2, candBuf, ctlTp);
  k_final   <<<gSlice, blk, 0, stream>>>(logits, V, chunk, ctlM, ctlT, ctlBp, ctlTp, out);
}